// GraphAttentionLayer_30090540876006
// MI455X (gfx1250) — compile-verified
//
#include <hip/hip_runtime.h>
#include <math.h>

typedef float v2f __attribute__((ext_vector_type(2)));
typedef float v8f __attribute__((ext_vector_type(8)));

#define DIN   64
#define DOUT  32
#define ALPHA 0.2f

// ---------------------------------------------------------------------------
// K0: init: zero out [N*32], zero denom [N], emax_enc [N] = INT_MIN
// ---------------------------------------------------------------------------
__global__ void gat_init_kernel(float* __restrict__ out, float* __restrict__ denom,
                                int* __restrict__ emax_enc, int N) {
    int i = blockIdx.x * blockDim.x + threadIdx.x;
    int total = N * DOUT;
    if (i < total) out[i] = 0.0f;
    if (i < N) {
        denom[i]    = 0.0f;
        emax_enc[i] = (int)0x80000000;  // decodes to non-finite -> becomes 0.0
    }
}

// ---------------------------------------------------------------------------
// K1: h = X @ W via V_WMMA_F32_16X16X4_F32.
// Block = 256 threads = 8 waves; each wave computes a 16-row x 32-col tile.
// W staged in LDS *transposed* (Wt[n*64 + k]) so each lane's B fragment
// (k, k+1 for fixed column n) is a contiguous, 8B-aligned ds_load_b64.
// A frag (16x4 f32, 2 VGPR): lane L -> row m = L%16, K = k0 + 2*(L/16) + {0,1}
// B frag (4x16 f32, 2 VGPR): lane L -> col n = L%16, K = k0 + 2*(L/16) + {0,1}
// C/D (16x16 f32, 8 VGPR):  VGPR v -> M = v + 8*(L/16), N = L%16
// ---------------------------------------------------------------------------
__global__ void gat_gemm_wmma_kernel(const float* __restrict__ X,
                                     const float* __restrict__ Wg,
                                     float* __restrict__ H, int N) {
    __shared__ float Wt[DOUT * DIN];  // 8 KB, transposed: Wt[n*DIN + k] = W[k*DOUT + n]
    int tid = threadIdx.x;
    #pragma unroll
    for (int i = tid; i < DIN * DOUT; i += 256) {
        int k = i / DOUT;
        int n = i % DOUT;
        Wt[n * DIN + k] = Wg[i];
    }
    __syncthreads();

    int wave = tid >> 5;
    int lane = tid & 31;
    int r0 = (blockIdx.x * 8 + wave) * 16;
    if (r0 >= N) return;  // uniform per-wave: EXEC stays all-1s on active waves

    int m    = lane & 15;          // A-row / B-col / C-col within tile
    int kp   = (lane >> 4) * 2;    // K sub-offset for upper half-wave
    int mrow = (lane >> 4) * 8;    // C/D M base for upper half-wave

    v8f c0 = {};
    v8f c1 = {};
    const float* arow = X + (size_t)(r0 + m) * DIN;
    const float* b0row = &Wt[(size_t)m * DIN];
    const float* b1row = &Wt[(size_t)(m + 16) * DIN];

    #pragma unroll
    for (int k0 = 0; k0 < DIN; k0 += 4) {
        v2f av = *(const v2f*)&arow[k0 + kp];   // 8B aligned: k0+kp even
        v2f b0 = *(const v2f*)&b0row[k0 + kp];  // contiguous in transposed LDS
        v2f b1 = *(const v2f*)&b1row[k0 + kp];
        // 8 args: (neg_a, A, neg_b, B, c_mod, C, reuse_a, reuse_b)
        c0 = __builtin_amdgcn_wmma_f32_16x16x4_f32(false, av, false, b0,
                                                   (short)0, c0, false, false);
        c1 = __builtin_amdgcn_wmma_f32_16x16x4_f32(false, av, false, b1,
                                                   (short)0, c1, false, false);
    }

    float* hbase = H + (size_t)r0 * DOUT;
    #pragma unroll
    for (int v = 0; v < 8; ++v) {
        int mm = mrow + v;
        hbase[(size_t)mm * DOUT + m]      = c0[v];
        hbase[(size_t)mm * DOUT + m + 16] = c1[v];
    }
}

// ---------------------------------------------------------------------------
// K2: per-node projections: sp[n] = h[n,:]·a[0:32], dp[n] = h[n,:]·a[32:64]
// ---------------------------------------------------------------------------
__global__ void gat_proj_kernel(const float* __restrict__ H,
                                const float* __restrict__ a,
                                float* __restrict__ sp, float* __restrict__ dp,
                                int N) {
    int n = blockIdx.x * blockDim.x + threadIdx.x;
    if (n >= N) return;
    const float* hr = H + (size_t)n * DOUT;
    float s = 0.0f, d = 0.0f;
    #pragma unroll
    for (int j = 0; j < DOUT; ++j) {
        float hv = hr[j];
        s = fmaf(hv, a[j], s);
        d = fmaf(hv, a[DOUT + j], d);
    }
    sp[n] = s;
    dp[n] = d;
}

// monotonic float -> signed-int key (order-preserving under signed compare)
__device__ __forceinline__ int enc_f32(float f) {
    int i = __float_as_int(f);
    return i >= 0 ? i : (int)(i ^ 0x7fffffff);
}

// ---------------------------------------------------------------------------
// K3: e = leaky_relu(sp[src] + dp[dst]); segment-max into emax_enc via atomicMax
// ---------------------------------------------------------------------------
__global__ void gat_edge_score_kernel(const float* __restrict__ sp,
                                      const float* __restrict__ dp,
                                      const int* __restrict__ src,
                                      const int* __restrict__ dst,
                                      float* __restrict__ e,
                                      int* __restrict__ emax_enc, int E) {
    int i = blockIdx.x * blockDim.x + threadIdx.x;
    if (i >= E) return;
    float f = sp[src[i]] + dp[dst[i]];
    f = fmaxf(f, ALPHA * f);  // leaky relu, alpha<1
    e[i] = f;
    atomicMax(&emax_enc[dst[i]], enc_f32(f));
}

// ---------------------------------------------------------------------------
// K4: decode emax; non-finite (no in-edges) -> 0.0 (matches reference where())
//     Writes float in-place over the int buffer.
// ---------------------------------------------------------------------------
__global__ void gat_max_finalize_kernel(int* __restrict__ emax_enc, int N) {
    int n = blockIdx.x * blockDim.x + threadIdx.x;
    if (n >= N) return;
    int k = emax_enc[n];
    float f = (k >= 0) ? __int_as_float(k) : __int_as_float(k ^ 0x7fffffff);
    if (!(fabsf(f) <= 3.4e38f)) f = 0.0f;  // NaN / +-inf -> 0
    emax_enc[n] = __float_as_int(f);
}

// ---------------------------------------------------------------------------
// K5: ex = exp(e - emax[dst]); denom[dst] += ex; e <- ex
// ---------------------------------------------------------------------------
__global__ void gat_edge_exp_kernel(float* __restrict__ e,
                                    const float* __restrict__ emax_f,
                                    const int* __restrict__ dst,
                                    float* __restrict__ denom, int E) {
    int i = blockIdx.x * blockDim.x + threadIdx.x;
    if (i >= E) return;
    int d = dst[i];
    float ex = __expf(e[i] - emax_f[d]);
    e[i] = ex;
    atomicAdd(&denom[d], ex);
}

// ---------------------------------------------------------------------------
// K6: one wave32 per edge, lane j = output column.
//     out[dst, j] += (e[i]/denom[dst]) * h[src, j]
//     Coalesced 128B gather of h[src]; 32 consecutive f32 atomics per wave.
// ---------------------------------------------------------------------------
__global__ void gat_aggregate_kernel(const float* __restrict__ e,
                                     const float* __restrict__ denom,
                                     const float* __restrict__ H,
                                     const int* __restrict__ src,
                                     const int* __restrict__ dst,
                                     float* __restrict__ out, int E) {
    int gid  = blockIdx.x * blockDim.x + threadIdx.x;
    int edge = gid >> 5;
    int j    = gid & 31;
    if (edge >= E) return;
    int s = src[edge];
    int d = dst[edge];
    float alpha = e[edge] / denom[d];
    atomicAdd(&out[(size_t)d * DOUT + j], alpha * H[(size_t)s * DOUT + j]);
}

// ---------------------------------------------------------------------------
// Launch
// inputs: 0 features [N*64] f32, 1 W [64*32] f32, 2 a [64] f32,
//         3 src [E] i32, 4 dst [E] i32
// output: [N*32] f32
// ---------------------------------------------------------------------------
extern "C" void kernel_launch(void* const* d_in, const int* in_sizes, int n_in,
                              void* d_out, int out_size, void* d_ws, size_t ws_size,
                              hipStream_t stream) {
    const float* X   = (const float*)d_in[0];
    const float* Wg  = (const float*)d_in[1];
    const float* a   = (const float*)d_in[2];
    const int*   src = (const int*)d_in[3];
    const int*   dst = (const int*)d_in[4];
    float*       out = (float*)d_out;

    const int N = in_sizes[0] / DIN;
    const int E = in_sizes[3];

    // workspace layout (all 4-byte elems): H | sp | dp | emax | denom | e
    float* H      = (float*)d_ws;
    float* sp     = H + (size_t)N * DOUT;
    float* dp     = sp + N;
    int*   emax   = (int*)(dp + N);
    float* denomv = (float*)(emax + N);
    float* ev     = denomv + N;

    const int T = 256;

    // K0: init accumulators
    {
        int total = N * DOUT;
        gat_init_kernel<<<(total + T - 1) / T, T, 0, stream>>>(out, denomv, emax, N);
    }
    // K1: WMMA GEMM (N is a multiple of 16; each block covers 128 rows)
    {
        int blocks = (N + 127) / 128;
        gat_gemm_wmma_kernel<<<blocks, T, 0, stream>>>(X, Wg, H, N);
    }
    // K2: projections
    gat_proj_kernel<<<(N + T - 1) / T, T, 0, stream>>>(H, a, sp, dp, N);
    // K3: edge scores + segment max
    gat_edge_score_kernel<<<(E + T - 1) / T, T, 0, stream>>>(sp, dp, src, dst, ev, emax, E);
    // K4: finalize max
    gat_max_finalize_kernel<<<(N + T - 1) / T, T, 0, stream>>>(emax, N);
    // K5: exp + denominator
    gat_edge_exp_kernel<<<(E + T - 1) / T, T, 0, stream>>>(ev, (const float*)emax, dst, denomv, E);
    // K6: weighted scatter aggregation (one wave per edge)
    {
        long long threads = (long long)E * 32;
        int blocks = (int)((threads + T - 1) / T);
        gat_aggregate_kernel<<<blocks, T, 0, stream>>>(ev, denomv, H, src, dst, out, E);
    }
}